// Feature_extraction_layer_basic_86526411145608
// MI455X (gfx1250) — compile-verified
//
#include <hip/hip_runtime.h>

#define FLT_MAX_F 3.402823466e+38f

typedef float v2f __attribute__((ext_vector_type(2)));
typedef float v8f __attribute__((ext_vector_type(8)));
typedef unsigned long long u64;

// Problem constants (fixed by setup_inputs)
#define BB 2
#define PP 16384
#define SS 4096
#define KK 16

__device__ __forceinline__ unsigned pack_key(float f) {
    unsigned ku = __float_as_uint(f);
    return ku ^ ((unsigned)((int)ku >> 31) | 0x80000000u);  // monotone float->u32
}
__device__ __forceinline__ u64 umin64(u64 a, u64 b) { return a < b ? a : b; }

// ---------------------------------------------------------------------------
// Kernel 1: per-batch min/max of channels 0,1 and pre-packed WMMA A-operands:
//   A0[p] = (-2*x, -2*y)            (lanes 0-15:  K=0,1)
//   A1[p] = (-2*z, x^2+y^2+z^2)     (lanes 16-31: K=2,3; 3D knn)
//   A2[p] = (x^2+y^2, 0)            (lanes 16-31: K=2,3; 2D knn)
// so key[m][n] = ||p_m||^2 - 2 p_m.q_n via A x (qx,qy,(qz|1),(1|0)).
// ---------------------------------------------------------------------------
__global__ void __launch_bounds__(256) k1_pack(const float* __restrict__ x,
                                               float* __restrict__ mnmx,
                                               float2* __restrict__ A0,
                                               float2* __restrict__ A1,
                                               float2* __restrict__ A2) {
    int b = blockIdx.x;
    int tid = threadIdx.x;
    const float* x0 = x + (size_t)b * 5 * PP;
    const float* x1 = x0 + PP;
    const float* x2 = x0 + 2 * PP;
    float mn0 = FLT_MAX_F, mn1 = FLT_MAX_F, mx0 = -FLT_MAX_F, mx1 = -FLT_MAX_F;
    for (int p = tid; p < PP; p += 256) {
        float a = x0[p], c1 = x1[p], c2 = x2[p];
        A0[(size_t)b * PP + p] = make_float2(-2.0f * a, -2.0f * c1);
        A1[(size_t)b * PP + p] = make_float2(-2.0f * c2, a * a + c1 * c1 + c2 * c2);
        A2[(size_t)b * PP + p] = make_float2(a * a + c1 * c1, 0.0f);
        mn0 = fminf(mn0, a);  mx0 = fmaxf(mx0, a);
        mn1 = fminf(mn1, c1); mx1 = fmaxf(mx1, c1);
    }
    __shared__ float red[4][256];
    red[0][tid] = mn0; red[1][tid] = mx0; red[2][tid] = mn1; red[3][tid] = mx1;
    __syncthreads();
    for (int s = 128; s > 0; s >>= 1) {
        if (tid < s) {
            red[0][tid] = fminf(red[0][tid], red[0][tid + s]);
            red[1][tid] = fmaxf(red[1][tid], red[1][tid + s]);
            red[2][tid] = fminf(red[2][tid], red[2][tid + s]);
            red[3][tid] = fmaxf(red[3][tid], red[3][tid + s]);
        }
        __syncthreads();
    }
    if (tid == 0) {
        mnmx[b * 4 + 0] = red[0][0];  // mn0
        mnmx[b * 4 + 1] = red[2][0];  // mn1
        mnmx[b * 4 + 2] = red[1][0];  // mx0
        mnmx[b * 4 + 3] = red[3][0];  // mx1
    }
}

// ---------------------------------------------------------------------------
// Kernel 2: 2D 1-NN of grid sample points via V_WMMA_F32_16X16X4_F32.
// One wave per 16 queries; two db tiles per iteration (clause-paired loads);
// per-tile candidates tree-reduced with v_min_u64 (packed key|index, exact
// ties); lane L / L+16 merged with shuffles.
// ---------------------------------------------------------------------------
__global__ void __launch_bounds__(256) k2_knn1(const float* __restrict__ x,
                                               const float* __restrict__ mnmx,
                                               const float* __restrict__ A0,
                                               const float* __restrict__ A2,
                                               float* __restrict__ qws) {
    int tid = threadIdx.x;
    int w = tid >> 5, lane = tid & 31;
    int gw = blockIdx.x * 8 + w;          // 0..511
    int b = gw >> 8;
    int s0 = (gw & 255) << 4;
    int L = lane & 15, half = lane >> 4;

    const v2f* Ap = (const v2f*)(half ? A2 : A0) + (size_t)b * PP + L;

    float mn0 = mnmx[b * 4 + 0], mn1 = mnmx[b * 4 + 1];
    float mx0 = mnmx[b * 4 + 2], mx1 = mnmx[b * 4 + 3];
    int m = s0 + L;
    float g0 = (float)(m & 63) * (1.0f / 64.0f);    // mesh[m % 64]
    float g1 = (float)(m >> 6) * (1.0f / 64.0f);    // mesh[m / 64]
    float q0 = g0 * (mx0 - mn0) + mn0;
    float q1 = g1 * (mx1 - mn1) + mn1;
    v2f bop;
    bop.x = half ? 1.0f : q0;
    bop.y = half ? 0.0f : q1;

    u64 best = ~0ull;
    for (int j0 = 0; j0 < PP; j0 += 32) {
        v2f a0 = Ap[j0];
        v2f a1 = Ap[j0 + 16];
        v8f z = {0.f, 0.f, 0.f, 0.f, 0.f, 0.f, 0.f, 0.f};
        v8f c0 = __builtin_amdgcn_wmma_f32_16x16x4_f32(false, a0, false, bop,
                                                       (short)0, z, false, false);
        v8f c1 = __builtin_amdgcn_wmma_f32_16x16x4_f32(false, a1, false, bop,
                                                       (short)0, z, false, false);
        // jb0/jb1 are 8-aligned, r<8: index low word formed with OR
        unsigned jb0 = (unsigned)(j0 + half * 8);
        unsigned jb1 = jb0 + 16u;
        u64 cd[16];
        #pragma unroll
        for (int r = 0; r < 8; ++r) {
            cd[r]     = ((u64)pack_key(c0[r]) << 32) | (u64)(jb0 | (unsigned)r);
            cd[r + 8] = ((u64)pack_key(c1[r]) << 32) | (u64)(jb1 | (unsigned)r);
        }
        // tree reduction (independent mins -> better ILP than a serial chain)
        #pragma unroll
        for (int i = 0; i < 8; ++i) cd[i] = umin64(cd[i], cd[i + 8]);
        #pragma unroll
        for (int i = 0; i < 4; ++i) cd[i] = umin64(cd[i], cd[i + 4]);
        cd[0] = umin64(cd[0], cd[2]);
        cd[1] = umin64(cd[1], cd[3]);
        best = umin64(best, umin64(cd[0], cd[1]));
    }
    // merge lane L with lane L+16 (packed key: tie -> lower db idx)
    unsigned lo = (unsigned)best, hi = (unsigned)(best >> 32);
    unsigned lo2 = __shfl_xor(lo, 16, 32);
    unsigned hi2 = __shfl_xor(hi, 16, 32);
    u64 other = ((u64)hi2 << 32) | (u64)lo2;
    best = umin64(best, other);
    if (lane < 16) {
        int bj = (int)(unsigned)best;
        const float* x0 = x + (size_t)b * 5 * PP;
        float4 qv;
        qv.x = x0[bj]; qv.y = x0[PP + bj]; qv.z = x0[2 * PP + bj]; qv.w = 0.0f;
        ((float4*)qws)[b * SS + s0 + L] = qv;
    }
}

// ---------------------------------------------------------------------------
// Kernel 3: 3D 16-NN via V_WMMA_F32_16X16X4_F32.
// Two db tiles per iteration (clause-paired loads). Fast path per candidate:
// pack (2 VALU) + one u32 threshold compare against wku = hi32(bk[15]);
// strict < is exact since stored indices always precede the current index.
// The guarded min/max chain is a no-op for non-qualifying candidates.
// Lane pair lists merged via LDS with exact tie handling.
// ---------------------------------------------------------------------------
__global__ void __launch_bounds__(256) k3_knn2(const float* __restrict__ A0,
                                               const float* __restrict__ A1,
                                               const float* __restrict__ qws,
                                               int* __restrict__ idx2) {
    __shared__ u64 mk[256][16];
    int tid = threadIdx.x;
    int w = tid >> 5, lane = tid & 31;
    int gw = blockIdx.x * 8 + w;          // 0..511
    int b = gw >> 8;
    int s0 = (gw & 255) << 4;
    int L = lane & 15, half = lane >> 4;

    const v2f* Ap = (const v2f*)(half ? A1 : A0) + (size_t)b * PP + L;

    const float4 qv = ((const float4*)qws)[b * SS + s0 + L];
    v2f bop;
    bop.x = half ? qv.z : qv.x;
    bop.y = half ? 1.0f : qv.y;

    u64 bk[16];
    #pragma unroll
    for (int i = 0; i < 16; ++i) bk[i] = ~0ull;
    unsigned wku = 0xFFFFFFFFu;

    for (int j0 = 0; j0 < PP; j0 += 32) {
        v2f a0 = Ap[j0];
        v2f a1 = Ap[j0 + 16];
        v8f z = {0.f, 0.f, 0.f, 0.f, 0.f, 0.f, 0.f, 0.f};
        v8f c0 = __builtin_amdgcn_wmma_f32_16x16x4_f32(false, a0, false, bop,
                                                       (short)0, z, false, false);
        v8f c1 = __builtin_amdgcn_wmma_f32_16x16x4_f32(false, a1, false, bop,
                                                       (short)0, z, false, false);
        unsigned jb = (unsigned)(j0 + half * 8);   // 8-aligned; low bits via OR
        #pragma unroll
        for (int t = 0; t < 16; ++t) {
            float key = (t < 8) ? c0[t & 7] : c1[t & 7];
            unsigned jj = (jb + (unsigned)(t < 8 ? 0 : 16)) | (unsigned)(t & 7);
            unsigned ku = pack_key(key);
            if (ku < wku) {
                u64 ck = ((u64)ku << 32) | (u64)jj;
                #pragma unroll
                for (int i = 0; i < 16; ++i) {
                    u64 old = bk[i];
                    u64 mn = ck < old ? ck : old;
                    ck      = ck < old ? old : ck;
                    bk[i] = mn;
                }
                wku = (unsigned)(bk[15] >> 32);
            }
        }
    }
    #pragma unroll
    for (int i = 0; i < 16; ++i) mk[tid][i] = bk[i];
    __syncthreads();
    if (lane < 16) {
        int s = s0 + L;
        int ia = 0, ib = 0;
        int* __restrict__ outp = idx2 + (size_t)b * 16 * SS + s;
        for (int o = 0; o < 16; ++o) {
            u64 va = mk[tid][ia], vb = mk[tid + 16][ib];
            bool ta = va < vb;              // packed key: tie -> lower db idx
            outp[(size_t)o * SS] = (int)(unsigned)(ta ? va : vb);
            if (ta) ++ia; else ++ib;
        }
    }
}

// ---------------------------------------------------------------------------
// Kernel 4: fused gather + PEmb + conv1x1s + attention + softmax + pooling.
// 128 threads = 8 s-positions x 16 neighbors. Writes pooled feature [B,128,S]
// to workspace and output rows 0..4.
// ---------------------------------------------------------------------------
__global__ void __launch_bounds__(128) k4_feature(const float* __restrict__ x,
                                                  const int* __restrict__ idx2,
                                                  const float* __restrict__ W_pos,
                                                  const float* __restrict__ b_pos,
                                                  const float* __restrict__ W_conv,
                                                  const float* __restrict__ b_conv,
                                                  const float* __restrict__ W_a1,
                                                  const float* __restrict__ b_a1,
                                                  const float* __restrict__ W_a2,
                                                  const float* __restrict__ b_a2,
                                                  float* __restrict__ featP,
                                                  float* __restrict__ out) {
    __shared__ float f5[8][16][5];
    __shared__ float feat[128][129];   // pad to avoid bank conflicts in pooling
    __shared__ float attv[128];
    __shared__ float wv[128];

    int t = threadIdx.x;
    int sl = t >> 4, kk = t & 15;
    int blk = blockIdx.x;              // 0..1023
    int b = blk >> 9;
    int s0 = (blk & 511) << 3;
    int s = s0 + sl;

    const float* xb = x + (size_t)b * 5 * PP;
    int j = idx2[((size_t)b * 16 + kk) * SS + s];
    #pragma unroll
    for (int c = 0; c < 5; ++c) f5[sl][kk][c] = xb[c * PP + j];
    __syncthreads();

    float pa0 = f5[sl][kk][0], pa1 = f5[sl][kk][1];
    float pa2 = f5[sl][kk][2], pa3 = f5[sl][kk][3];
    float pc0 = f5[sl][0][0], pc1 = f5[sl][0][1], pc2 = f5[sl][0][2];
    float d0 = pa0 - pc0, d1 = pa1 - pc1, d2 = pa2 - pc2;
    float temp = sqrtf(d0 * d0 + d1 * d1 + d2 * d2 + 1e-12f);
    float pemb[10] = {pa0, pa1, pa2, pc0, pc1, pc2, d0, d1, d2, temp};

    // channels 0..63: W_conv on x1[:, :4]
    #pragma unroll 4
    for (int c = 0; c < 64; ++c) {
        const float* wr = W_conv + c * 4;
        feat[c][t] = b_conv[c] + wr[0] * pa0 + wr[1] * pa1 + wr[2] * pa2 + wr[3] * pa3;
    }
    // channels 64..127: W_pos on PEmb
    #pragma unroll 2
    for (int c = 0; c < 64; ++c) {
        const float* wr = W_pos + c * 10;
        float acc = b_pos[c];
        #pragma unroll
        for (int i = 0; i < 10; ++i) acc += wr[i] * pemb[i];
        feat[64 + c][t] = acc;
    }
    __syncthreads();

    // attention: att = W_a2 . leaky_relu(W_a1 . feat + b_a1, 0.2) + b_a2
    float h[32];
    #pragma unroll
    for (int i = 0; i < 32; ++i) h[i] = b_a1[i];
    for (int c = 0; c < 128; ++c) {
        float f = feat[c][t];
        #pragma unroll
        for (int i = 0; i < 32; ++i) h[i] += W_a1[i * 128 + c] * f;
    }
    float att = b_a2[0];
    #pragma unroll
    for (int i = 0; i < 32; ++i) {
        float hv = h[i];
        hv = hv > 0.0f ? hv : 0.2f * hv;
        att += W_a2[i] * hv;
    }
    attv[t] = att;
    __syncthreads();

    // softmax over k within each s
    float mx = -FLT_MAX_F;
    #pragma unroll
    for (int i = 0; i < 16; ++i) mx = fmaxf(mx, attv[sl * 16 + i]);
    float den = 0.0f;
    #pragma unroll
    for (int i = 0; i < 16; ++i) den += expf(attv[sl * 16 + i] - mx);
    wv[t] = expf(att - mx) / den;
    __syncthreads();

    // pooled[c][s] = sum_k feat[c][s,k] * w[s,k]; thread t handles channel c=t
    float* fP = featP + (size_t)b * 128 * SS;
    #pragma unroll
    for (int si = 0; si < 8; ++si) {
        float acc = 0.0f;
        #pragma unroll
        for (int k = 0; k < 16; ++k) acc += feat[t][si * 16 + k] * wv[si * 16 + k];
        fP[(size_t)t * SS + s0 + si] = acc;
    }

    // output rows 0..4 = x1[:, :5, s, 0]
    if (kk == 0) {
        float* ob = out + (size_t)b * 133 * SS;
        #pragma unroll
        for (int c = 0; c < 5; ++c) ob[c * SS + s] = f5[sl][0][c];
    }
}

// ---------------------------------------------------------------------------
// Kernel 5: out[5+m][n] = W_b1[m][:] . featP[:][n] + b_b1[m] via
// V_WMMA_F32_16X16X4_F32. One wave per 16x16 output tile, K=128 in 32 steps,
// C initialized with the bias (broadcast along N).
// ---------------------------------------------------------------------------
__global__ void __launch_bounds__(256) k5_gemm(const float* __restrict__ W_b1,
                                               const float* __restrict__ b_b1,
                                               const float* __restrict__ featP,
                                               float* __restrict__ out) {
    int tid = threadIdx.x;
    int w = tid >> 5, lane = tid & 31;
    int gw = blockIdx.x * 8 + w;       // 0..4095
    int b = gw >> 11;
    int rem = gw & 2047;
    int m0 = (rem >> 8) << 4;          // 8 m-tiles
    int n0 = (rem & 255) << 4;         // 256 n-tiles
    int L = lane & 15, half = lane >> 4;

    const float* fP = featP + (size_t)b * 128 * SS;
    const float* wr = W_b1 + (size_t)(m0 + L) * 128;

    v8f c;
    #pragma unroll
    for (int r = 0; r < 8; ++r) c[r] = b_b1[m0 + half * 8 + r];

    #pragma unroll 4
    for (int k0 = 0; k0 < 128; k0 += 4) {
        int kr = k0 + half * 2;
        v2f a = *(const v2f*)(wr + kr);
        v2f bb;
        bb.x = fP[(size_t)kr * SS + n0 + L];
        bb.y = fP[(size_t)(kr + 1) * SS + n0 + L];
        c = __builtin_amdgcn_wmma_f32_16x16x4_f32(false, a, false, bb,
                                                  (short)0, c, false, false);
    }

    float* ob = out + (size_t)b * 133 * SS + (size_t)(5 + m0 + half * 8) * SS + n0 + L;
    #pragma unroll
    for (int r = 0; r < 8; ++r) ob[(size_t)r * SS] = c[r];
}

// ---------------------------------------------------------------------------
extern "C" void kernel_launch(void* const* d_in, const int* in_sizes, int n_in,
                              void* d_out, int out_size, void* d_ws, size_t ws_size,
                              hipStream_t stream) {
    const float* x      = (const float*)d_in[0];
    // d_in[1] = num_points (4096, compile-time constant here)
    const float* W_pos  = (const float*)d_in[2];
    const float* b_pos  = (const float*)d_in[3];
    const float* W_conv = (const float*)d_in[4];
    const float* b_conv = (const float*)d_in[5];
    const float* W_a1   = (const float*)d_in[6];
    const float* b_a1   = (const float*)d_in[7];
    const float* W_a2   = (const float*)d_in[8];
    const float* b_a2   = (const float*)d_in[9];
    const float* W_b1   = (const float*)d_in[10];
    const float* b_b1   = (const float*)d_in[11];
    float* out = (float*)d_out;

    // Workspace layout
    char* ws = (char*)d_ws;
    float*  mnmx  = (float*) (ws + 0);        // 8 floats
    float2* A0    = (float2*)(ws + 256);      // 2*16384 float2 = 256 KB
    float2* A1    = (float2*)(ws + 262912);   // 256 KB
    float2* A2    = (float2*)(ws + 525568);   // 256 KB
    float*  qws   = (float*) (ws + 788224);   // 2*4096 float4 (16-aligned)
    int*    idx2  = (int*)   (ws + 919552);   // 2*16*4096 ints
    float*  featP = (float*) (ws + 1444096);  // 2*128*4096 floats

    k1_pack   <<<dim3(2),    dim3(256), 0, stream>>>(x, mnmx, A0, A1, A2);
    k2_knn1   <<<dim3(64),   dim3(256), 0, stream>>>(x, mnmx, (const float*)A0,
                                                     (const float*)A2, qws);
    k3_knn2   <<<dim3(64),   dim3(256), 0, stream>>>((const float*)A0, (const float*)A1,
                                                     qws, idx2);
    k4_feature<<<dim3(1024), dim3(128), 0, stream>>>(x, idx2, W_pos, b_pos,
                                                     W_conv, b_conv, W_a1, b_a1,
                                                     W_a2, b_a2, featP, out);
    k5_gemm   <<<dim3(512),  dim3(256), 0, stream>>>(W_b1, b_b1, featP, out);
}